// HistoricalEdgeModule_6305011991195
// MI455X (gfx1250) — compile-verified
//
#include <hip/hip_runtime.h>
#include <hip/hip_bf16.h>
#include <math.h>

typedef float v2f __attribute__((ext_vector_type(2)));
typedef float v8f __attribute__((ext_vector_type(8)));

#define N_NODES  100000
#define N_PAIRS  500000
#define N_EVENTS 1000000
#define N_TYPES  4
#define EMBED    128

// ---------------------------------------------------------------------------
// Kernel 1: typed projection via fp32 WMMA.
// Each block = one tile of 16 nodes; 8 waves = 8 column tiles of 16.
// Compute all 4 type projections with V_WMMA_F32_16X16X4_F32, then select the
// per-row result branchlessly (v_cndmask) to avoid divergent EXEC branches.
// ---------------------------------------------------------------------------
__global__ void __launch_bounds__(256)
proj_kernel(const float* __restrict__ E, const float* __restrict__ W,
            const int* __restrict__ types, float* __restrict__ proj) {
    const int node0 = blockIdx.x * 16;
    const int wave  = threadIdx.x >> 5;     // 0..7 -> column tile
    const int lane  = threadIdx.x & 31;
    const int e0    = wave * 16;
    const int half  = lane >> 4;            // 0: K pair base 0, 1: base 2
    const int l15   = lane & 15;

    // A: lane holds row (l&15), floats [k0+2*half, k0+2*half+1]
    const float* aptr = E + (size_t)(node0 + l15) * EMBED + 2 * half;
    // B: lane holds col e0+(l&15); W[t][e][d] contiguous in d
    const float* bptr = W + (size_t)(e0 + l15) * EMBED + 2 * half;
    const size_t tstride = (size_t)EMBED * EMBED;

    v8f c0 = {}, c1 = {}, c2 = {}, c3 = {};

#pragma unroll 4
    for (int k0 = 0; k0 < EMBED; k0 += 4) {
        v2f a  = *(const v2f*)(aptr + k0);
        v2f b0 = *(const v2f*)(bptr + k0);
        v2f b1 = *(const v2f*)(bptr + tstride + k0);
        v2f b2 = *(const v2f*)(bptr + 2 * tstride + k0);
        v2f b3 = *(const v2f*)(bptr + 3 * tstride + k0);
        c0 = __builtin_amdgcn_wmma_f32_16x16x4_f32(false, a, false, b0, (short)0, c0, false, false);
        c1 = __builtin_amdgcn_wmma_f32_16x16x4_f32(false, a, false, b1, (short)0, c1, false, false);
        c2 = __builtin_amdgcn_wmma_f32_16x16x4_f32(false, a, false, b2, (short)0, c2, false, false);
        c3 = __builtin_amdgcn_wmma_f32_16x16x4_f32(false, a, false, b3, (short)0, c3, false, false);
    }

    // C/D layout: VGPR r, lanes 0-15 -> row r, lanes 16-31 -> row r+8; col = e0+(l&15)
    // Branchless 4-way select per row (lowers to v_cndmask_b32, no divergence).
#pragma unroll
    for (int r = 0; r < 8; ++r) {
        const int rowg = node0 + r + 8 * half;
        const int t = types[rowg];
        const float lo = (t == 1) ? c1[r] : c0[r];
        const float hi = (t == 3) ? c3[r] : c2[r];
        const float v  = (t >= 2) ? hi : lo;
        proj[(size_t)rowg * EMBED + e0 + l15] = v;
    }
}

// ---------------------------------------------------------------------------
// Kernel 2a: per-block partial sums of exp(-theta*(T - t_i))  (deterministic)
// ---------------------------------------------------------------------------
__global__ void __launch_bounds__(256)
event_partial_kernel(const float* __restrict__ times, const float* __restrict__ curT,
                     const float* __restrict__ theta, float* __restrict__ partials) {
    __shared__ float red[256];
    const float T  = curT[0];
    const float th = theta[0];
    float s = 0.f;
    for (int i = blockIdx.x * blockDim.x + threadIdx.x; i < N_EVENTS;
         i += gridDim.x * blockDim.x) {
        s += __expf(-th * (T - times[i]));
    }
    red[threadIdx.x] = s;
    __syncthreads();
#pragma unroll
    for (int off = 128; off > 0; off >>= 1) {
        if (threadIdx.x < off) red[threadIdx.x] += red[threadIdx.x + off];
        __syncthreads();
    }
    if (threadIdx.x == 0) partials[blockIdx.x] = red[0];
}

// ---------------------------------------------------------------------------
// Kernel 2b: final tree reduction of 1024 partials -> scalar sum_alpha
// ---------------------------------------------------------------------------
__global__ void __launch_bounds__(1024)
event_final_kernel(const float* __restrict__ partials, float* __restrict__ out_scalar) {
    __shared__ float red[1024];
    red[threadIdx.x] = partials[threadIdx.x];
    __syncthreads();
#pragma unroll
    for (int off = 512; off > 0; off >>= 1) {
        if (threadIdx.x < off) red[threadIdx.x] += red[threadIdx.x + off];
        __syncthreads();
    }
    if (threadIdx.x == 0) out_scalar[0] = red[0];
}

// ---------------------------------------------------------------------------
// Kernel 3: per-pair gamma = -||proj[m]-proj[n]|| + sum_alpha
// One wave (32 lanes) per pair; each lane handles a float4 (128 = 32*4).
// proj table (51.2MB) is L2-resident, so the 512MB of gathers hit L2.
// ---------------------------------------------------------------------------
__global__ void __launch_bounds__(256)
pair_kernel(const int* __restrict__ pairs, const float* __restrict__ proj,
            const float* __restrict__ sum_alpha_p, float* __restrict__ out) {
    const int pid  = (blockIdx.x * blockDim.x + threadIdx.x) >> 5;
    const int lane = threadIdx.x & 31;
    if (pid >= N_PAIRS) return;

    const int m = pairs[2 * pid];
    const int n = pairs[2 * pid + 1];
    const float4 a = ((const float4*)(proj + (size_t)m * EMBED))[lane];
    const float4 b = ((const float4*)(proj + (size_t)n * EMBED))[lane];
    float dx = a.x - b.x, dy = a.y - b.y, dz = a.z - b.z, dw = a.w - b.w;
    float s = dx * dx + dy * dy + dz * dz + dw * dw;
#pragma unroll
    for (int off = 16; off > 0; off >>= 1) s += __shfl_xor(s, off, 32);
    if (lane == 0) out[pid] = -sqrtf(s) + sum_alpha_p[0];
}

// ---------------------------------------------------------------------------
extern "C" void kernel_launch(void* const* d_in, const int* in_sizes, int n_in,
                              void* d_out, int out_size, void* d_ws, size_t ws_size,
                              hipStream_t stream) {
    const int*   pairs  = (const int*)d_in[0];
    const float* embeds = (const float*)d_in[1];
    const int*   types  = (const int*)d_in[2];
    const float* etimes = (const float*)d_in[3];
    const float* curT   = (const float*)d_in[4];
    const float* W      = (const float*)d_in[5];
    const float* theta  = (const float*)d_in[6];
    float* out = (float*)d_out;

    // workspace layout: proj table | 1024 partials | sum_alpha scalar
    float* proj      = (float*)d_ws;
    float* partials  = proj + (size_t)N_NODES * EMBED;
    float* sum_alpha = partials + 1024;

    proj_kernel<<<N_NODES / 16, 256, 0, stream>>>(embeds, W, types, proj);
    event_partial_kernel<<<1024, 256, 0, stream>>>(etimes, curT, theta, partials);
    event_final_kernel<<<1, 1024, 0, stream>>>(partials, sum_alpha);
    pair_kernel<<<N_PAIRS / 8, 256, 0, stream>>>(pairs, proj, sum_alpha, out);
}